// PR2d_13391708029804
// MI455X (gfx1250) — compile-verified
//
#include <hip/hip_runtime.h>

// Problem constants (match reference)
#define NB    4
#define NCIN  20
#define NCOUT 20
#define NM1   4
#define NM2   4
#define NYY   64
#define NXX   64
#define PI2   6.28318530717958647692f

typedef __attribute__((ext_vector_type(2))) float v2f;
typedef __attribute__((ext_vector_type(8))) float v8f;

static __device__ __forceinline__ v8f wmma_f32(v2f a, v2f b, v8f c) {
  // D = A(16x4) * B(4x16) + C(16x16), fp32, wave32
  return __builtin_amdgcn_wmma_f32_16x16x4_f32(false, a, false, b, (short)0, c, false, false);
}

static __device__ __forceinline__ float2 cfma(float2 a, float2 b, float2 acc) {
  acc.x += a.x * b.x - a.y * b.y;
  acc.y += a.x * b.y + a.y * b.x;
  return acc;
}

// Fast twiddle: angle = sign * 2*pi*t/64, t in [0,64). Uses v_sin_f32/v_cos_f32.
static __device__ __forceinline__ void twiddle(int t, float sign, float* s, float* c) {
  const float ang = sign * (PI2 / 64.0f) * (float)t;
  *s = __sinf(ang);
  *c = __cosf(ang);
}

// ---------------------------------------------------------------------------
// Kernel 1: DFT along x (real input -> complex).  C = X(5120x64) * F(64x64)
// grid=320 blocks (16 rows each), block=128 (4 waves; wave = 16-col tile)
// ---------------------------------------------------------------------------
__global__ __launch_bounds__(128) void k_fft_x(const float* __restrict__ xin,
                                               float2* __restrict__ out) {
  const int lane = threadIdx.x & 31;
  const int wave = threadIdx.x >> 5;
  const int lm = lane & 15;
  const int lh = lane >> 4;
  const int row0 = blockIdx.x * 16;   // 5120 rows total
  const int col  = wave * 16 + lm;    // output frequency column (N)
  const float* xr = xin + (size_t)(row0 + lm) * NXX;

  v8f acc_re = {}; v8f acc_im = {};
  for (int k0 = 0; k0 < NXX; k0 += 4) {
    const int kb = k0 + 2 * lh;
    v2f a; a.x = xr[kb]; a.y = xr[kb + 1];
    float s0, c0, s1, c1;
    twiddle((kb * col) & 63, -1.0f, &s0, &c0);
    twiddle(((kb + 1) * col) & 63, -1.0f, &s1, &c1);
    v2f br; br.x = c0; br.y = c1;
    v2f bi; bi.x = s0; bi.y = s1;
    acc_re = wmma_f32(a, br, acc_re);
    acc_im = wmma_f32(a, bi, acc_im);
  }
  for (int j = 0; j < 8; ++j) {
    const int m = j + 8 * lh;
    out[(size_t)(row0 + m) * NXX + col] = make_float2(acc_re[j], acc_im[j]);
  }
}

// ---------------------------------------------------------------------------
// Kernel 2: DFT along y (complex).  out[o][xf] = sum_y W(o*y) * in[y][xf]
// grid=(4 o-tiles, 80 planes), block=128 (wave = xf tile)
// ---------------------------------------------------------------------------
__global__ __launch_bounds__(128) void k_fft_y(const float2* __restrict__ in,
                                               float2* __restrict__ out) {
  const int lane = threadIdx.x & 31;
  const int wave = threadIdx.x >> 5;
  const int lm = lane & 15;
  const int lh = lane >> 4;
  const float2* src = in + (size_t)blockIdx.y * (NYY * NXX);
  float2* dst = out + (size_t)blockIdx.y * (NYY * NXX);
  const int o0 = blockIdx.x * 16;
  const int col = wave * 16 + lm;
  const int orow = o0 + lm;

  v8f rr = {}, ri = {}, ir = {}, ii = {};
  for (int k0 = 0; k0 < NYY; k0 += 4) {
    const int kb = k0 + 2 * lh;
    float s0, c0, s1, c1;
    twiddle((orow * kb) & 63, -1.0f, &s0, &c0);
    twiddle((orow * (kb + 1)) & 63, -1.0f, &s1, &c1);
    v2f ar; ar.x = c0; ar.y = c1;   // twiddle on A side (M=o, K=y)
    v2f ai; ai.x = s0; ai.y = s1;
    const float2 b0 = src[kb * NXX + col];
    const float2 b1 = src[(kb + 1) * NXX + col];
    v2f br; br.x = b0.x; br.y = b1.x;
    v2f bi; bi.x = b0.y; bi.y = b1.y;
    rr = wmma_f32(ar, br, rr);
    ii = wmma_f32(ai, bi, ii);
    ri = wmma_f32(ar, bi, ri);
    ir = wmma_f32(ai, br, ir);
  }
  for (int j = 0; j < 8; ++j) {
    const int m = j + 8 * lh;
    dst[(o0 + m) * NXX + col] = make_float2(rr[j] - ii[j], ri[j] + ir[j]);
  }
}

// ---------------------------------------------------------------------------
// Kernel 3: r1[o,i,k,p] = 1/(lam1[o]-wp1[i,k,p]), r2[xf,i,k,q] analogous
// ---------------------------------------------------------------------------
__global__ __launch_bounds__(256) void k_poles(const float2* __restrict__ wp1,
                                               const float2* __restrict__ wp2,
                                               const float* __restrict__ tx,
                                               const float* __restrict__ ty,
                                               float2* __restrict__ r1,
                                               float2* __restrict__ r2) {
  const int idx = blockIdx.x * blockDim.x + threadIdx.x;  // 2*64*20*20*4 = 204800
  const int half = idx / (64 * NCIN * NCOUT * 4);
  const int r = idx % (64 * NCIN * NCOUT * 4);
  const int p = r & 3;
  const int k = (r >> 2) % NCOUT;
  const int i = ((r >> 2) / NCOUT) % NCIN;
  const int o = r / (4 * NCOUT * NCIN);
  const int f = (o < 32) ? o : o - 64;
  const float* t = half ? tx : ty;
  const float dt = t[1] - t[0];
  const float lam = PI2 * ((float)f / 64.0f) / dt;  // Im(lambda); Re = 0
  const float2 w = (half ? wp2 : wp1)[(i * NCOUT + k) * 4 + p];
  const float dre = -w.x;
  const float dim = lam - w.y;
  const float inv = 1.0f / (dre * dre + dim * dim);
  (half ? r2 : r1)[r] = make_float2(dre * inv, -dim * inv);
}

// ---------------------------------------------------------------------------
// Kernel 4: e1[c,i,p,z] = exp(wp1[c,i,p]*ty[z]); e2[c,i,q,x] analogous
// ---------------------------------------------------------------------------
__global__ __launch_bounds__(256) void k_exps(const float2* __restrict__ wp1,
                                              const float2* __restrict__ wp2,
                                              const float* __restrict__ tx,
                                              const float* __restrict__ ty,
                                              float2* __restrict__ e1,
                                              float2* __restrict__ e2) {
  const int idx = blockIdx.x * blockDim.x + threadIdx.x;  // 2*20*20*4*64 = 204800
  const int half = idx / (NCIN * NCOUT * 4 * 64);
  const int r = idx % (NCIN * NCOUT * 4 * 64);
  const int z = r & 63;
  const int p = (r >> 6) & 3;
  const int i = (r >> 8) % NCOUT;
  const int c = r / (64 * 4 * NCOUT);
  const float t = (half ? tx : ty)[z];
  const float2 w = (half ? wp2 : wp1)[(c * NCOUT + i) * 4 + p];
  const float mag = __expf(w.x * t);
  const float s = __sinf(w.y * t);
  const float co = __cosf(w.y * t);
  (half ? e2 : e1)[r] = make_float2(mag * co, mag * s);
}

// ---------------------------------------------------------------------------
// Kernel 5: A1[o,i,k,q] = sum_p wres[i,k,p,q] * r1[o,i,k,p]
// ---------------------------------------------------------------------------
__global__ __launch_bounds__(256) void k_A1(const float2* __restrict__ wres,
                                            const float2* __restrict__ r1,
                                            float2* __restrict__ A1) {
  const int idx = blockIdx.x * blockDim.x + threadIdx.x;  // 64*20*20*4 = 102400
  const int q = idx & 3;
  const int k = (idx >> 2) % NCOUT;
  const int i = ((idx >> 2) / NCOUT) % NCIN;
  const int o = idx / (4 * NCOUT * NCIN);
  float2 acc = make_float2(0.0f, 0.0f);
  for (int p = 0; p < 4; ++p)
    acc = cfma(wres[((i * NCOUT + k) * 4 + p) * 4 + q],
               r1[((o * NCIN + i) * NCOUT + k) * 4 + p], acc);
  A1[idx] = acc;
}

// ---------------------------------------------------------------------------
// Kernel 6: out_res1[b,k,o,xf] = sum_i alpha[b,i,o,xf] * (sum_q A1*r2)
// ---------------------------------------------------------------------------
__global__ __launch_bounds__(256) void k_outres1(const float2* __restrict__ alpha,
                                                 const float2* __restrict__ A1,
                                                 const float2* __restrict__ r2,
                                                 float2* __restrict__ res1) {
  const int idx = blockIdx.x * blockDim.x + threadIdx.x;  // 4*20*64*64 = 327680
  const int xf = idx & 63;
  const int o = (idx >> 6) & 63;
  const int k = (idx >> 12) % NCOUT;
  const int b = idx / (4096 * NCOUT);
  float2 acc = make_float2(0.0f, 0.0f);
  for (int i = 0; i < NCIN; ++i) {
    const float2* a1p = A1 + ((o * NCIN + i) * NCOUT + k) * 4;
    const float2* r2p = r2 + ((xf * NCIN + i) * NCOUT + k) * 4;
    float2 g = make_float2(0.0f, 0.0f);
    for (int q = 0; q < 4; ++q) g = cfma(a1p[q], r2p[q], g);
    acc = cfma(alpha[((b * NCIN + i) * NYY + o) * NXX + xf], g, acc);
  }
  res1[((b * NCOUT + k) * NYY + o) * NXX + xf] = acc;
}

// ---------------------------------------------------------------------------
// Kernel 7: IFFT along xf (complex): tmp[o][x] = (1/64) sum_xf res1[o][xf] W+(xf*x)
// ---------------------------------------------------------------------------
__global__ __launch_bounds__(128) void k_ifft_x(const float2* __restrict__ in,
                                                float2* __restrict__ out) {
  const int lane = threadIdx.x & 31;
  const int wave = threadIdx.x >> 5;
  const int lm = lane & 15;
  const int lh = lane >> 4;
  const float2* src = in + (size_t)blockIdx.y * (NYY * NXX);
  float2* dst = out + (size_t)blockIdx.y * (NYY * NXX);
  const int o0 = blockIdx.x * 16;
  const int col = wave * 16 + lm;
  const float2* arow = src + (o0 + lm) * NXX;

  v8f rr = {}, ri = {}, ir = {}, ii = {};
  for (int k0 = 0; k0 < NXX; k0 += 4) {
    const int kb = k0 + 2 * lh;
    const float2 a0 = arow[kb];
    const float2 a1 = arow[kb + 1];
    v2f ar; ar.x = a0.x; ar.y = a1.x;   // data on A side (M=o, K=xf)
    v2f ai; ai.x = a0.y; ai.y = a1.y;
    float s0, c0, s1, c1;
    twiddle((kb * col) & 63, 1.0f, &s0, &c0);
    twiddle(((kb + 1) * col) & 63, 1.0f, &s1, &c1);
    v2f br; br.x = c0; br.y = c1;
    v2f bi; bi.x = s0; bi.y = s1;
    rr = wmma_f32(ar, br, rr);
    ii = wmma_f32(ai, bi, ii);
    ri = wmma_f32(ar, bi, ri);
    ir = wmma_f32(ai, br, ir);
  }
  const float sc = 1.0f / 64.0f;
  for (int j = 0; j < 8; ++j) {
    const int m = j + 8 * lh;
    dst[(o0 + m) * NXX + col] = make_float2((rr[j] - ii[j]) * sc, (ri[j] + ir[j]) * sc);
  }
}

// ---------------------------------------------------------------------------
// Kernel 8: IFFT along o, real part only, writes x1 to d_out
// ---------------------------------------------------------------------------
__global__ __launch_bounds__(128) void k_ifft_y_real(const float2* __restrict__ in,
                                                     float* __restrict__ out) {
  const int lane = threadIdx.x & 31;
  const int wave = threadIdx.x >> 5;
  const int lm = lane & 15;
  const int lh = lane >> 4;
  const float2* src = in + (size_t)blockIdx.y * (NYY * NXX);
  float* dst = out + (size_t)blockIdx.y * (NYY * NXX);
  const int z0 = blockIdx.x * 16;
  const int col = wave * 16 + lm;
  const int zrow = z0 + lm;

  v8f rr = {}, ii = {};
  for (int k0 = 0; k0 < NYY; k0 += 4) {
    const int kb = k0 + 2 * lh;
    float s0, c0, s1, c1;
    twiddle((zrow * kb) & 63, 1.0f, &s0, &c0);
    twiddle((zrow * (kb + 1)) & 63, 1.0f, &s1, &c1);
    v2f ar; ar.x = c0; ar.y = c1;   // twiddle on A side (M=z, K=o)
    v2f ai; ai.x = s0; ai.y = s1;
    const float2 b0 = src[kb * NXX + col];
    const float2 b1 = src[(kb + 1) * NXX + col];
    v2f br; br.x = b0.x; br.y = b1.x;
    v2f bi; bi.x = b0.y; bi.y = b1.y;
    rr = wmma_f32(ar, br, rr);   // Re = Ar*Br - Ai*Bi
    ii = wmma_f32(ai, bi, ii);
  }
  const float sc = 1.0f / 64.0f;
  for (int j = 0; j < 8; ++j) {
    const int m = j + 8 * lh;
    dst[(z0 + m) * NXX + col] = (rr[j] - ii[j]) * sc;
  }
}

// ---------------------------------------------------------------------------
// Kernel 9: T[b,i,k,o,q] = sum_xf alpha[b,i,o,xf] * r2[xf,i,k,q]
// ---------------------------------------------------------------------------
__global__ __launch_bounds__(256) void k_T(const float2* __restrict__ alpha,
                                           const float2* __restrict__ r2,
                                           float2* __restrict__ T) {
  const int idx = blockIdx.x * blockDim.x + threadIdx.x;  // 4*20*20*64*4 = 409600
  const int q = idx & 3;
  const int o = (idx >> 2) & 63;
  const int k = (idx >> 8) % NCOUT;
  const int i = (idx / 5120) % NCIN;
  const int b = idx / 102400;
  const float2* al = alpha + ((b * NCIN + i) * NYY + o) * NXX;
  float2 acc = make_float2(0.0f, 0.0f);
  for (int xf = 0; xf < 64; ++xf)
    acc = cfma(al[xf], r2[((xf * NCIN + i) * NCOUT + k) * 4 + q], acc);
  T[idx] = acc;
}

// ---------------------------------------------------------------------------
// Kernel 10: S[b,i,k,p,q] = sum_o r1[o,i,k,p] * T[b,i,k,o,q]
// ---------------------------------------------------------------------------
__global__ __launch_bounds__(256) void k_S(const float2* __restrict__ r1,
                                           const float2* __restrict__ T,
                                           float2* __restrict__ S) {
  const int idx = blockIdx.x * blockDim.x + threadIdx.x;  // 4*20*20*4*4 = 25600
  const int q = idx & 3;
  const int p = (idx >> 2) & 3;
  const int k = (idx >> 4) % NCOUT;
  const int i = (idx / 320) % NCIN;
  const int b = idx / 6400;
  float2 acc = make_float2(0.0f, 0.0f);
  for (int o = 0; o < 64; ++o)
    acc = cfma(r1[((o * NCIN + i) * NCOUT + k) * 4 + p],
               T[(((b * NCIN + i) * NCOUT + k) * 64 + o) * 4 + q], acc);
  S[idx] = acc;
}

// ---------------------------------------------------------------------------
// Kernel 11: out_res2[b,k,p,q] = sum_i wres[i,k,p,q] * S[b,i,k,p,q]
// ---------------------------------------------------------------------------
__global__ __launch_bounds__(256) void k_outres2(const float2* __restrict__ wres,
                                                 const float2* __restrict__ S,
                                                 float2* __restrict__ o2) {
  const int idx = blockIdx.x * blockDim.x + threadIdx.x;  // 4*20*4*4 = 1280
  const int q = idx & 3;
  const int p = (idx >> 2) & 3;
  const int k = (idx >> 4) % NCOUT;
  const int b = idx / 320;
  float2 acc = make_float2(0.0f, 0.0f);
  for (int i = 0; i < NCIN; ++i)
    acc = cfma(wres[((i * NCOUT + k) * 4 + p) * 4 + q],
               S[(((b * NCIN + i) * NCOUT + k) * 4 + p) * 4 + q], acc);
  o2[idx] = acc;
}

// ---------------------------------------------------------------------------
// Kernel 12: U[b,i,c,q,z] = sum_p out_res2[b,c,p,q] * e1[c,i,p,z]
// ---------------------------------------------------------------------------
__global__ __launch_bounds__(256) void k_U(const float2* __restrict__ o2,
                                           const float2* __restrict__ e1,
                                           float2* __restrict__ U) {
  const int idx = blockIdx.x * blockDim.x + threadIdx.x;  // 4*20*20*4*64 = 409600
  const int z = idx & 63;
  const int q = (idx >> 6) & 3;
  const int c = (idx >> 8) % NCOUT;
  const int i = (idx / 5120) % NCOUT;
  const int b = idx / 102400;
  float2 acc = make_float2(0.0f, 0.0f);
  for (int p = 0; p < 4; ++p)
    acc = cfma(o2[((b * NCOUT + c) * 4 + p) * 4 + q],
               e1[((c * NCOUT + i) * 4 + p) * 64 + z], acc);
  U[idx] = acc;
}

// ---------------------------------------------------------------------------
// Kernel 13: out[b,i,z,x] += Re(sum_{c,q} U[b,i,c,q,z]*e2[c,i,q,x]) / 4096
// ---------------------------------------------------------------------------
__global__ __launch_bounds__(256) void k_x2add(const float2* __restrict__ U,
                                               const float2* __restrict__ e2,
                                               float* __restrict__ out) {
  const int idx = blockIdx.x * blockDim.x + threadIdx.x;  // 4*20*64*64 = 327680
  const int x = idx & 63;
  const int z = (idx >> 6) & 63;
  const int i = (idx >> 12) % NCOUT;
  const int b = idx / (4096 * NCOUT);
  float acc = 0.0f;
  for (int c = 0; c < NCOUT; ++c) {
    const float2* up = U + ((((b * NCOUT + i) * NCOUT + c) * 4) * 64) + z;
    const float2* ep = e2 + ((c * NCOUT + i) * 4) * 64 + x;
    for (int q = 0; q < 4; ++q) {
      const float2 u = up[q * 64];
      const float2 e = ep[q * 64];
      acc += u.x * e.x - u.y * e.y;
    }
  }
  out[idx] += acc * (1.0f / 4096.0f);
}

// ---------------------------------------------------------------------------
extern "C" void kernel_launch(void* const* d_in, const int* in_sizes, int n_in,
                              void* d_out, int out_size, void* d_ws, size_t ws_size,
                              hipStream_t stream) {
  (void)in_sizes; (void)n_in; (void)out_size; (void)ws_size;

  const float*  x    = (const float*)d_in[0];
  const float2* wp1  = (const float2*)d_in[1];   // complex64 (20,20,4)
  const float2* wp2  = (const float2*)d_in[2];   // complex64 (20,20,4)
  const float2* wres = (const float2*)d_in[3];   // complex64 (20,20,4,4)
  const float*  tx   = (const float*)d_in[4];
  const float*  ty   = (const float*)d_in[5];
  float* out = (float*)d_out;

  // Workspace layout (float2 slots), ~21.4 MB total
  char* ws = (char*)d_ws;
  float2* alphaX = (float2*)ws; ws += (size_t)327680 * 8;  // row-FFT result
  float2* alpha  = (float2*)ws; ws += (size_t)327680 * 8;  // full fft2
  float2* r1     = (float2*)ws; ws += (size_t)102400 * 8;
  float2* r2     = (float2*)ws; ws += (size_t)102400 * 8;
  float2* e1     = (float2*)ws; ws += (size_t)102400 * 8;
  float2* e2     = (float2*)ws; ws += (size_t)102400 * 8;
  float2* A1     = (float2*)ws; ws += (size_t)102400 * 8;
  float2* res1   = (float2*)ws; ws += (size_t)327680 * 8;
  float2* tmpI   = (float2*)ws; ws += (size_t)327680 * 8;
  float2* Tb     = (float2*)ws; ws += (size_t)409600 * 8;
  float2* Sb     = (float2*)ws; ws += (size_t)25600 * 8;
  float2* o2     = (float2*)ws; ws += (size_t)1280 * 8;
  float2* Ub     = (float2*)ws; ws += (size_t)409600 * 8;

  // Spectral path
  k_fft_x<<<320, 128, 0, stream>>>(x, alphaX);                       // WMMA
  k_fft_y<<<dim3(4, 80), 128, 0, stream>>>(alphaX, alpha);           // WMMA
  k_poles<<<800, 256, 0, stream>>>(wp1, wp2, tx, ty, r1, r2);
  k_exps<<<800, 256, 0, stream>>>(wp1, wp2, tx, ty, e1, e2);
  k_A1<<<400, 256, 0, stream>>>(wres, r1, A1);
  k_outres1<<<1280, 256, 0, stream>>>(alpha, A1, r2, res1);
  k_ifft_x<<<dim3(4, 80), 128, 0, stream>>>(res1, tmpI);             // WMMA
  k_ifft_y_real<<<dim3(4, 80), 128, 0, stream>>>(tmpI, out);         // WMMA, writes x1
  // Pole-expansion path
  k_T<<<1600, 256, 0, stream>>>(alpha, r2, Tb);
  k_S<<<100, 256, 0, stream>>>(r1, Tb, Sb);
  k_outres2<<<5, 256, 0, stream>>>(wres, Sb, o2);
  k_U<<<1600, 256, 0, stream>>>(o2, e1, Ub);
  k_x2add<<<1280, 256, 0, stream>>>(Ub, e2, out);                    // out += x2
}